// REDecoder_58213986730277
// MI455X (gfx1250) — compile-verified
//
#include <hip/hip_runtime.h>
#include <hip/hip_bf16.h>
#include <math.h>

// ---------------------------------------------------------------------------
// Problem constants (from reference): B=2, L=512, H=768, E=512, R=32768, H2=384
// ---------------------------------------------------------------------------
constexpr int B  = 2;
constexpr int L  = 512;
constexpr int H  = 768;
constexpr int E  = 512;
constexpr int R  = 32768;
constexpr int H2 = 384;
constexpr int NROWS = B * E;          // 1024 unique entity rows
constexpr int TWOH  = 2 * H;          // 1536

// Workspace layout (in floats)
constexpr size_t OFF_X  = 0;                                  // [1024,1536]
constexpr size_t OFF_YH = OFF_X  + (size_t)NROWS * TWOH;      // [1024,768]
constexpr size_t OFF_YT = OFF_YH + (size_t)NROWS * H;
constexpr size_t OFF_HH = OFF_YT + (size_t)NROWS * H;         // [1024,384]
constexpr size_t OFF_HT = OFF_HH + (size_t)NROWS * H2;
constexpr size_t OFF_U  = OFF_HT + (size_t)NROWS * H2;        // [2,1024,384]
constexpr size_t OFF_LH = OFF_U  + (size_t)2 * NROWS * H2;    // [1024,2]
constexpr size_t OFF_LT = OFF_LH + (size_t)NROWS * 2;         // [1024,2]
constexpr size_t OFF_P  = OFF_LT + (size_t)NROWS * 2;         // [8192] CE partials

typedef __attribute__((ext_vector_type(2))) float v2f;
typedef __attribute__((ext_vector_type(8))) float v8f;

// ---------------------------------------------------------------------------
// Kernel 1: gather per-entity features  X[row] = [hidden[b,start], emb[label]]
// float4 path: 768 floats = 192 float4 per half-row.
// ---------------------------------------------------------------------------
__global__ void build_features(const float* __restrict__ hs,
                               const int* __restrict__ ent_start,
                               const int* __restrict__ ent_label,
                               const float* __restrict__ emb,
                               float* __restrict__ X)
{
    const int row = blockIdx.x;            // b*E + e
    const int b   = row / E;
    const int tok = ent_start[row];
    const int lab = ent_label[row];
    const float4* __restrict__ s0 = (const float4*)(hs  + ((size_t)b * L + tok) * H);
    const float4* __restrict__ s1 = (const float4*)(emb + (size_t)lab * H);
    float4* __restrict__ d0 = (float4*)(X + (size_t)row * TWOH);
    float4* __restrict__ d1 = d0 + (H / 4);
    for (int i = threadIdx.x; i < H / 4; i += blockDim.x) {
        d0[i] = s0[i];
        d1[i] = s1[i];
    }
}

// ---------------------------------------------------------------------------
// Kernel 2: fp32 WMMA GEMM, dual problem select by blockIdx.z:
//   C[z] = act(A[z][M,K] @ W[z][K,N] + bias[z])
// Block = 256 threads (8 waves). Wave w -> 16 rows (m0 = bx*128 + w*16),
// each wave computes a 16x64 N-strip (4 accumulators, A fragment reused 4x).
// V_WMMA_F32_16X16X4_F32 fragment layout (ISA 7.12.2):
//   A (16x4): lanes 0-15 hold K=0(v0),1(v1); lanes 16-31 hold K=2(v0),3(v1)
//   B (4x16): row striped across lanes; v0: K=0 (lo)/K=2 (hi), v1: K=1/K=3
//   C/D: element (M = v + 8*half, N = lane&15)
// z-select is wave-uniform (scalar cselect): EXEC stays all-ones for WMMA.
// Grid exactly tiles M,N (all dims multiples of 16/64/128) -> no bounds checks.
// ---------------------------------------------------------------------------
__global__ void gemm_wmma_f32(const float* __restrict__ A0, const float* __restrict__ A1,
                              const float* __restrict__ W0, const float* __restrict__ W1,
                              const float* __restrict__ b0, const float* __restrict__ b1,
                              float* __restrict__ C0, float* __restrict__ C1,
                              int K, int N, int doRelu)
{
    const int z = blockIdx.z;
    const float* __restrict__ A    = z ? A1 : A0;
    const float* __restrict__ Wz   = z ? W1 : W0;
    const float* __restrict__ bias = z ? b1 : b0;
    float* __restrict__ Cz         = z ? C1 : C0;

    const int wave = threadIdx.x >> 5;
    const int lane = threadIdx.x & 31;
    const int half = lane >> 4;            // 0: lanes 0-15, 1: lanes 16-31
    const int lid  = lane & 15;

    const int m0 = blockIdx.x * 128 + wave * 16;
    const int n0 = blockIdx.y * 64;

    v8f acc[4] = {v8f{}, v8f{}, v8f{}, v8f{}};

    const float* __restrict__ Arow = A + (size_t)(m0 + lid) * K;

    for (int k = 0; k < K; k += 4) {
        v2f a;
        a.x = Arow[k + 2 * half + 0];
        a.y = Arow[k + 2 * half + 1];
        const float* __restrict__ Wk0 = Wz + (size_t)(k + 2 * half + 0) * N;
        const float* __restrict__ Wk1 = Wz + (size_t)(k + 2 * half + 1) * N;
#pragma unroll
        for (int t = 0; t < 4; ++t) {
            const int n = n0 + t * 16 + lid;
            v2f bfrag;
            bfrag.x = Wk0[n];
            bfrag.y = Wk1[n];
            acc[t] = __builtin_amdgcn_wmma_f32_16x16x4_f32(
                /*neg_a=*/false, a, /*neg_b=*/false, bfrag,
                /*c_mod=*/(short)0, acc[t], /*reuse_a=*/false, /*reuse_b=*/false);
        }
    }

#pragma unroll
    for (int t = 0; t < 4; ++t) {
        const int n = n0 + t * 16 + lid;
        const float bv = bias ? bias[n] : 0.0f;
#pragma unroll
        for (int v = 0; v < 8; ++v) {
            const int m = m0 + v + 8 * half;
            float val = acc[t][v] + bv;
            if (doRelu) val = fmaxf(val, 0.0f);
            Cz[(size_t)m * N + n] = val;
        }
    }
}

// ---------------------------------------------------------------------------
// Kernel 3: per-entity linear terms  linH[row,o] = heads[row]·lin_W[0:384,o],
//                                    linT[row,o] = tails[row]·lin_W[384:768,o]
// One wave per row; 4 waves per block.
// ---------------------------------------------------------------------------
__global__ void lin_proj(const float* __restrict__ Hh, const float* __restrict__ Ht,
                         const float* __restrict__ linW,
                         float* __restrict__ linH, float* __restrict__ linT)
{
    const int wave = threadIdx.x >> 5;
    const int lane = threadIdx.x & 31;
    const int row  = blockIdx.x * 4 + wave;
    const float* __restrict__ h = Hh + (size_t)row * H2;
    const float* __restrict__ t = Ht + (size_t)row * H2;
    float sh0 = 0.f, sh1 = 0.f, st0 = 0.f, st1 = 0.f;
    for (int i = lane; i < H2; i += 32) {
        const float hv = h[i], tv = t[i];
        sh0 += hv * linW[(size_t)i * 2 + 0];
        sh1 += hv * linW[(size_t)i * 2 + 1];
        st0 += tv * linW[(size_t)(H2 + i) * 2 + 0];
        st1 += tv * linW[(size_t)(H2 + i) * 2 + 1];
    }
#pragma unroll
    for (int off = 16; off > 0; off >>= 1) {
        sh0 += __shfl_down(sh0, off);
        sh1 += __shfl_down(sh1, off);
        st0 += __shfl_down(st0, off);
        st1 += __shfl_down(st1, off);
    }
    if (lane == 0) {
        linH[(size_t)row * 2 + 0] = sh0;
        linH[(size_t)row * 2 + 1] = sh1;
        linT[(size_t)row * 2 + 0] = st0;
        linT[(size_t)row * 2 + 1] = st1;
    }
}

// ---------------------------------------------------------------------------
// Kernel 4: per-relation biaffine logits + CE. One wave per relation (8/block).
// logits[o] = U[o][hrow]·tails[trow] + linH[hrow,o] + linT[trow,o] + lin_b[o]
// float4 loads: 384 floats = 96 float4 = 32 lanes x 3. All b128, coalesced.
// Per-block CE partial sums written to ws (deterministic final reduction).
// ---------------------------------------------------------------------------
__global__ void relation_kernel(const int* __restrict__ rel_head,
                                const int* __restrict__ rel_tail,
                                const int* __restrict__ rel_label,
                                const float* __restrict__ U,
                                const float* __restrict__ Ht,
                                const float* __restrict__ linH,
                                const float* __restrict__ linT,
                                const float* __restrict__ lin_b,
                                float* __restrict__ out_logits,   // d_out + 1
                                float* __restrict__ partials)
{
    __shared__ float sCE[8];
    const int wave = threadIdx.x >> 5;
    const int lane = threadIdx.x & 31;
    const int idx  = blockIdx.x * 8 + wave;   // b*R + r, 0..65535
    const int b    = idx >> 15;               // R = 32768
    const int hrow = b * E + rel_head[idx];
    const int trow = b * E + rel_tail[idx];
    const int lab  = rel_label[idx];

    const float4* __restrict__ u0 = (const float4*)(U + (size_t)hrow * H2);
    const float4* __restrict__ u1 = (const float4*)(U + (size_t)(NROWS + hrow) * H2);
    const float4* __restrict__ tv = (const float4*)(Ht + (size_t)trow * H2);

    float s0 = 0.f, s1 = 0.f;
#pragma unroll
    for (int j = 0; j < 3; ++j) {
        const int i = lane + 32 * j;          // 96 float4s per row
        const float4 x = tv[i];
        const float4 a = u0[i];
        const float4 c = u1[i];
        s0 += a.x * x.x + a.y * x.y + a.z * x.z + a.w * x.w;
        s1 += c.x * x.x + c.y * x.y + c.z * x.z + c.w * x.w;
    }
#pragma unroll
    for (int off = 16; off > 0; off >>= 1) {
        s0 += __shfl_down(s0, off);
        s1 += __shfl_down(s1, off);
    }
    if (lane == 0) {
        const float l0 = s0 + linH[(size_t)hrow * 2 + 0] + linT[(size_t)trow * 2 + 0] + lin_b[0];
        const float l1 = s1 + linH[(size_t)hrow * 2 + 1] + linT[(size_t)trow * 2 + 1] + lin_b[1];
        out_logits[(size_t)idx * 2 + 0] = l0;
        out_logits[(size_t)idx * 2 + 1] = l1;
        const float m   = fmaxf(l0, l1);
        const float lse = m + logf(expf(l0 - m) + expf(l1 - m));
        sCE[wave] = lse - (lab == 0 ? l0 : l1);
    }
    __syncthreads();
    if (threadIdx.x == 0) {
        float acc = 0.f;
#pragma unroll
        for (int w = 0; w < 8; ++w) acc += sCE[w];
        partials[blockIdx.x] = acc;
    }
}

// ---------------------------------------------------------------------------
// Kernel 5: deterministic loss reduction.  loss = (sum of partials) / R
// ---------------------------------------------------------------------------
__global__ void reduce_loss(const float* __restrict__ partials, float* __restrict__ loss,
                            int n, float scale)
{
    __shared__ float sm[256];
    float acc = 0.f;
    for (int i = threadIdx.x; i < n; i += 256) acc += partials[i];
    sm[threadIdx.x] = acc;
    __syncthreads();
    for (int s = 128; s > 0; s >>= 1) {
        if ((int)threadIdx.x < s) sm[threadIdx.x] += sm[threadIdx.x + s];
        __syncthreads();
    }
    if (threadIdx.x == 0) *loss = sm[0] * scale;
}

// ---------------------------------------------------------------------------
extern "C" void kernel_launch(void* const* d_in, const int* in_sizes, int n_in,
                              void* d_out, int out_size, void* d_ws, size_t ws_size,
                              hipStream_t stream)
{
    const float* hs        = (const float*)d_in[0];   // [B,L,H]
    const int*   ent_start = (const int*)  d_in[1];   // [B,E]
    const int*   ent_label = (const int*)  d_in[2];   // [B,E]
    const int*   rel_head  = (const int*)  d_in[3];   // [B,R]
    const int*   rel_tail  = (const int*)  d_in[4];   // [B,R]
    const int*   rel_label = (const int*)  d_in[5];   // [B,R]
    const float* emb       = (const float*)d_in[6];   // [3,H]
    const float* head_W1   = (const float*)d_in[7];   // [2H,H]
    const float* head_b1   = (const float*)d_in[8];   // [H]
    const float* head_W2   = (const float*)d_in[9];   // [H,H2]
    const float* head_b2   = (const float*)d_in[10];  // [H2]
    const float* tail_W1   = (const float*)d_in[11];
    const float* tail_b1   = (const float*)d_in[12];
    const float* tail_W2   = (const float*)d_in[13];
    const float* tail_b2   = (const float*)d_in[14];
    const float* bil_W     = (const float*)d_in[15];  // [2,H2,H2]
    const float* lin_W     = (const float*)d_in[16];  // [2*H2,2]
    const float* lin_b     = (const float*)d_in[17];  // [2]

    float* ws = (float*)d_ws;
    float* X    = ws + OFF_X;
    float* Yh   = ws + OFF_YH;
    float* Yt   = ws + OFF_YT;
    float* Hh   = ws + OFF_HH;
    float* Htl  = ws + OFF_HT;
    float* U    = ws + OFF_U;
    float* linH = ws + OFF_LH;
    float* linT = ws + OFF_LT;
    float* part = ws + OFF_P;

    float* out = (float*)d_out;   // out[0]=loss, out[1..]=logits [B*R,2]

    // 1) gather unique-entity features
    build_features<<<NROWS, 256, 0, stream>>>(hs, ent_start, ent_label, emb, X);

    // 2) MLP layer 1 (head & tail fused via z): [1024,1536] x [1536,768] -> ReLU
    {
        dim3 grid(NROWS / 128, H / 64, 2);
        gemm_wmma_f32<<<grid, 256, 0, stream>>>(X, X, head_W1, tail_W1,
                                                head_b1, tail_b1, Yh, Yt,
                                                TWOH, H, 1);
    }
    // 3) MLP layer 2 (head & tail fused via z): [1024,768] x [768,384] -> ReLU
    {
        dim3 grid(NROWS / 128, H2 / 64, 2);
        gemm_wmma_f32<<<grid, 256, 0, stream>>>(Yh, Yt, head_W2, tail_W2,
                                                head_b2, tail_b2, Hh, Htl,
                                                H, H2, 1);
    }
    // 4) bilinear precompute (o=0,1 fused via z): U[o] = Hh @ bil_W[o]
    {
        dim3 grid(NROWS / 128, H2 / 64, 2);
        gemm_wmma_f32<<<grid, 256, 0, stream>>>(Hh, Hh, bil_W, bil_W + (size_t)H2 * H2,
                                                nullptr, nullptr, U, U + (size_t)NROWS * H2,
                                                H2, H2, 0);
    }
    // 5) per-entity linear terms
    lin_proj<<<NROWS / 4, 128, 0, stream>>>(Hh, Htl, lin_W, linH, linT);

    // 6) relations: logits + per-block CE partials
    relation_kernel<<<(B * R) / 8, 256, 0, stream>>>(rel_head, rel_tail, rel_label,
                                                     U, Htl, linH, linT, lin_b,
                                                     out + 1, part);
    // 7) deterministic loss reduction
    reduce_loss<<<1, 256, 0, stream>>>(part, out, (B * R) / 8, 1.0f / (float)R);
}